// EncoderMaskClassifier_34823594836192
// MI455X (gfx1250) — compile-verified
//
#include <hip/hip_runtime.h>
#include <math.h>

// ---------------------------------------------------------------------------
// Problem constants (from the reference)
// ---------------------------------------------------------------------------
#define NB      256      // N samples
#define CAND    4
#define CHW     27648    // C*H*W = 3*96*96
#define HH      96
#define WW      96
#define PATCH   32
#define NOISEC  2
#define FD      512
#define NPIX    (PATCH * PATCH)     // 1024 crop positions

#define KSPLIT1 12                  // K-split for the big GEMM (wave-count!)
#define KCHUNK1 (CHW / KSPLIT1)     // 2304, multiple of 4

typedef __attribute__((ext_vector_type(2))) float v2f;
typedef __attribute__((ext_vector_type(8))) float v8f;

__device__ __forceinline__ float gelu_exact(float v) {
    // 0.5 * v * (1 + erf(v / sqrt(2)))  -- jax.nn.gelu(approximate=False)
    return 0.5f * v * (1.0f + erff(v * 0.70710678118654752440f));
}

__device__ __forceinline__ v8f wmma4(v2f a, v2f b, v8f c) {
    return __builtin_amdgcn_wmma_f32_16x16x4_f32(
        false, a, false, b, (short)0, c, false, false);
}

// ---------------------------------------------------------------------------
// fp32 WMMA partial GEMM:  P[z][M,N] = A[M, k0:k0+Kchunk] @ B[k0:k0+Kchunk, N]
//
// One wave computes a 32Mx32N register tile (2x2 WMMA tiles, 32 accum VGPRs):
// each A fragment feeds 2 WMMAs, each B fragment feeds 2 WMMAs -> 1.5 loads
// per WMMA and 2x less L2 re-read than a 16x16 wave tile. Block = 4 waves in
// a 2x2 arrangement (64Mx64N macro-tile) so the WGP$ absorbs intra-block
// re-reads. gridDim.z splits K so the launch has enough waves to cover HBM
// latency at 23.3 TB/s; partials are summed in a deterministic reduce pass.
// Inner loop uses incremented base pointers + immediate offsets to minimize
// per-iteration address VALU.
// ---------------------------------------------------------------------------
__global__ __launch_bounds__(128)
void wmma_gemm_part(const float* __restrict__ A, const float* __restrict__ B,
                    float* __restrict__ P, int M, int K, int N, int Kchunk)
{
    const int lane  = threadIdx.x & 31;
    const int wave  = threadIdx.x >> 5;
    const int half  = lane >> 4;        // 0: lanes 0-15, 1: lanes 16-31
    const int l16   = lane & 15;
    const int waveM = wave >> 1;
    const int waveN = wave & 1;
    const int m0    = blockIdx.y * 64 + waveM * 32;
    const int n0    = blockIdx.x * 64 + waveN * 32;
    const int k0    = blockIdx.z * Kchunk;

    v8f c00 = {0.f,0.f,0.f,0.f,0.f,0.f,0.f,0.f};
    v8f c01 = c00, c10 = c00, c11 = c00;

    // A 16x4 fp32 fragment: lane holds A[m=l16, kk + {0,1}], kk = k + 2*half
    // B 4x16 fp32 fragment: lane holds B[kk + {0,1}, n=l16]
    const float* __restrict__ pa0 = A + (size_t)(m0 + l16) * K + k0 + 2 * half;
    const float* __restrict__ pa1 = pa0 + (size_t)16 * K;
    const float* __restrict__ pb  = B + (size_t)(k0 + 2 * half) * N + n0 + l16;

    #pragma unroll 4
    for (int k = 0; k < Kchunk; k += 4) {
        v2f a0 = *(const v2f*)pa0;           // contiguous pair -> b64 load
        v2f a1 = *(const v2f*)pa1;
        v2f b0, b1;
        b0[0] = pb[0];   b0[1] = pb[N];      // rows kk, kk+1 ; imm offsets
        b1[0] = pb[16];  b1[1] = pb[N + 16];
        pa0 += 4; pa1 += 4; pb += (size_t)4 * N;
        c00 = wmma4(a0, b0, c00);
        c01 = wmma4(a0, b1, c01);
        c10 = wmma4(a1, b0, c10);
        c11 = wmma4(a1, b1, c11);
    }

    float* __restrict__ Pz = P + (size_t)blockIdx.z * M * N;
    // C/D layout: VGPR v of lane -> row = v + 8*half, col = l16
    #pragma unroll
    for (int v = 0; v < 8; ++v) {
        const int r = m0 + v + 8 * half;
        Pz[(size_t)r        * N + n0      + l16] = c00[v];
        Pz[(size_t)r        * N + n0 + 16 + l16] = c01[v];
        Pz[(size_t)(r + 16) * N + n0      + l16] = c10[v];
        Pz[(size_t)(r + 16) * N + n0 + 16 + l16] = c11[v];
    }
}

// ---------------------------------------------------------------------------
// Deterministic K-split reduction + optional bias + optional exact-GELU.
//   Out[i] = act( sum_z P[z][i] + bias[i % N] )
// ---------------------------------------------------------------------------
__global__ __launch_bounds__(256)
void reduce_bias_act(const float* __restrict__ P, float* __restrict__ Out,
                     const float* __restrict__ bias, int total, int N, int S,
                     int apply_gelu)
{
    const int i = blockIdx.x * 256 + threadIdx.x;
    if (i >= total) return;
    float s = 0.0f;
    for (int z = 0; z < S; ++z) s += P[(size_t)z * total + i];
    if (bias) s += bias[i & (N - 1)];      // N is a power of two (512)
    if (apply_gelu) s = gelu_exact(s);
    Out[i] = s;
}

// ---------------------------------------------------------------------------
// Per-sample crop gather-GEMM + fused z assembly + gelu.
// One block per sample n, 256 threads.
//   cropSum[c,f] = sum_p v_c[p] * W_enc[r0(p), f]       (1024 gathered rows)
//   z[c,f]      = 2*b_enc[f] + Z0[n,f] + cropSum[c,f] - cropSum[0,f]
//   Zg[n*4+c,f] = gelu(z)
// ---------------------------------------------------------------------------
__global__ __launch_bounds__(256)
void crop_encode_kernel(const float* __restrict__ state,
                        const int* __restrict__ x, const int* __restrict__ y,
                        const int* __restrict__ noise_x, const int* __restrict__ noise_y,
                        const float* __restrict__ W_enc, const float* __restrict__ b_enc,
                        const float* __restrict__ Z0, float* __restrict__ Zg)
{
    __shared__ int    r0s[NPIX];
    __shared__ float4 v4s[NPIX];

    const int n   = blockIdx.x;
    const int tid = threadIdx.x;
    const int xn  = x[n];
    const int yn  = y[n];

    int nxs[3], nys[3];
    #pragma unroll
    for (int c = 0; c < 3; ++c) {
        int nx = xn + noise_x[c * NB + n] - NOISEC;
        int ny = yn + noise_y[c * NB + n] - NOISEC;
        nxs[c] = min(max(nx, 0), WW - PATCH - 1);   // clip to [0, 63]
        nys[c] = min(max(ny, 0), HH - PATCH - 1);
    }

    const float* __restrict__ sn = state + (size_t)n * CHW;

    // Stage gather rows + all 4 candidate values (float4 -> b128 DS traffic).
    for (int p = tid; p < NPIX; p += 256) {
        const int px = p >> 5;
        const int py = p & 31;
        const int r0 = (yn + py) + HH * (xn + px);  // faithful H-stride quirk
        r0s[p] = r0;
        float4 v;
        v.x = sn[r0];                               // s0 value (pre-zero state)
        float* vp = &v.x;
        #pragma unroll
        for (int c = 0; c < 3; ++c) {
            const int ix = nxs[c] + px;
            const int iy = nys[c] + py;
            // gathered from the zeroed state: zero if inside the cut rect
            const bool inside = (ix >= xn) && (ix < xn + PATCH) &&
                                (iy >= yn) && (iy < yn + PATCH);
            vp[c + 1] = inside ? 0.0f : sn[iy + HH * ix];
        }
        v4s[p] = v;
    }
    __syncthreads();

    // Each thread owns 2 feature columns.
    const int f0 = tid;
    const int f1 = tid + 256;
    float a00 = 0.f, a01 = 0.f, a10 = 0.f, a11 = 0.f;
    float a20 = 0.f, a21 = 0.f, a30 = 0.f, a31 = 0.f;

    #pragma unroll 4
    for (int p = 0; p < NPIX; ++p) {
        const int r = r0s[p];
        const float w0 = W_enc[(size_t)r * FD + f0];
        const float w1 = W_enc[(size_t)r * FD + f1];
        const float4 v = v4s[p];
        a00 = fmaf(v.x, w0, a00);  a01 = fmaf(v.x, w1, a01);
        a10 = fmaf(v.y, w0, a10);  a11 = fmaf(v.y, w1, a11);
        a20 = fmaf(v.z, w0, a20);  a21 = fmaf(v.z, w1, a21);
        a30 = fmaf(v.w, w0, a30);  a31 = fmaf(v.w, w1, a31);
    }

    const float base0 = 2.0f * b_enc[f0] + Z0[(size_t)n * FD + f0];
    const float base1 = 2.0f * b_enc[f1] + Z0[(size_t)n * FD + f1];
    const float d0[CAND] = {0.f, a10 - a00, a20 - a00, a30 - a00};
    const float d1[CAND] = {0.f, a11 - a01, a21 - a01, a31 - a01};

    #pragma unroll
    for (int c = 0; c < CAND; ++c) {
        const size_t q = (size_t)(n * CAND + c);
        Zg[q * FD + f0] = gelu_exact(base0 + d0[c]);
        Zg[q * FD + f1] = gelu_exact(base1 + d1[c]);
    }
}

// ---------------------------------------------------------------------------
// Head: logit[q] = Hg[q,:] . W2 + b2, then apply the candidate shuffle and
// emit (logits, targets). One block per sample; wave c handles candidate c
// (wave32 shuffle reduction). Thread 0 does the 4-element stable argsort.
// ---------------------------------------------------------------------------
__global__ __launch_bounds__(128)
void head_kernel(const float* __restrict__ Hg, const float* __restrict__ W2,
                 const float* __restrict__ b2, const float* __restrict__ shuffle,
                 float* __restrict__ out)
{
    __shared__ float logit_s[CAND];
    const int n    = blockIdx.x;
    const int tid  = threadIdx.x;
    const int wave = tid >> 5;
    const int lane = tid & 31;

    const float* __restrict__ hrow = Hg + (size_t)(n * CAND + wave) * FD;
    float partial = 0.0f;
    #pragma unroll 4
    for (int j = lane; j < FD; j += 32)
        partial = fmaf(hrow[j], W2[j], partial);
    #pragma unroll
    for (int off = 16; off > 0; off >>= 1)
        partial += __shfl_xor(partial, off, 32);
    if (lane == 0) logit_s[wave] = partial + b2[0];
    __syncthreads();

    if (tid == 0) {
        float sv[CAND];
        #pragma unroll
        for (int c = 0; c < CAND; ++c) sv[c] = shuffle[n * CAND + c];
        bool used[CAND] = {false, false, false, false};
        int target = 0;
        #pragma unroll
        for (int j = 0; j < CAND; ++j) {        // stable ascending argsort
            int best = -1; float bv = 0.0f;
            #pragma unroll
            for (int i = 0; i < CAND; ++i) {
                if (used[i]) continue;
                if (best < 0 || sv[i] < bv) { best = i; bv = sv[i]; }
            }
            used[best] = true;
            out[n * CAND + j] = logit_s[best];  // logits of shuffled candidates
            if (best == 0) target = j;          // true crop landed at slot j
        }
        out[NB * CAND + n] = (float)target;     // targets after the logits
    }
}

// ---------------------------------------------------------------------------
// Launch
// ---------------------------------------------------------------------------
extern "C" void kernel_launch(void* const* d_in, const int* in_sizes, int n_in,
                              void* d_out, int out_size, void* d_ws, size_t ws_size,
                              hipStream_t stream)
{
    const float* state   = (const float*)d_in[0];
    const int*   x       = (const int*)  d_in[1];
    const int*   y       = (const int*)  d_in[2];
    const int*   noise_x = (const int*)  d_in[3];
    const int*   noise_y = (const int*)  d_in[4];
    const float* shuffle = (const float*)d_in[5];
    const float* W_enc   = (const float*)d_in[6];
    const float* b_enc   = (const float*)d_in[7];
    const float* W1      = (const float*)d_in[8];
    const float* b1      = (const float*)d_in[9];
    const float* W2      = (const float*)d_in[10];
    const float* b2      = (const float*)d_in[11];
    float* out = (float*)d_out;

    float* Z0 = (float*)d_ws;                  // 256  x 512
    float* Zg = Z0 + (size_t)NB * FD;          // 1024 x 512  gelu(z)
    float* Hg = Zg + (size_t)NB * CAND * FD;   // 1024 x 512  gelu(h)
    float* P  = Hg + (size_t)NB * CAND * FD;   // K-split partials (max 6 MB)

    // 1) P[z] = state @ W_enc (chunked K), then Z0 = sum_z P[z]
    //    M=256, K=27648, N=512: grid (8,4,12) = 384 blocks, 1536 waves.
    wmma_gemm_part<<<dim3(FD / 64, NB / 64, KSPLIT1), 128, 0, stream>>>(
        state, W_enc, P, NB, CHW, FD, KCHUNK1);
    reduce_bias_act<<<(NB * FD + 255) / 256, 256, 0, stream>>>(
        P, Z0, nullptr, NB * FD, FD, KSPLIT1, 0);

    // 2) crop gather-GEMM + z assembly + gelu -> Zg
    crop_encode_kernel<<<NB, 256, 0, stream>>>(
        state, x, y, noise_x, noise_y, W_enc, b_enc, Z0, Zg);

    // 3) Hg = gelu(Zg @ W1 + b1)   M=1024, K=512, N=512: grid (8,16,1)
    wmma_gemm_part<<<dim3(FD / 64, (NB * CAND) / 64, 1), 128, 0, stream>>>(
        Zg, W1, P, NB * CAND, FD, FD, FD);
    reduce_bias_act<<<(NB * CAND * FD + 255) / 256, 256, 0, stream>>>(
        P, Hg, b1, NB * CAND * FD, FD, 1, 1);

    // 4) logits + shuffle + targets
    head_kernel<<<NB, 128, 0, stream>>>(Hg, W2, b2, shuffle, out);
}